// EdgeConv_80874234184112
// MI455X (gfx1250) — compile-verified
//
#include <hip/hip_runtime.h>
#include <hip/hip_bf16.h>

// EdgeConv fused kernels for gfx1250 (MI455X), wave32, WMMA f32 16x16x4.

typedef __attribute__((ext_vector_type(2))) float v2f;
typedef __attribute__((ext_vector_type(8))) float v8f;
typedef __attribute__((ext_vector_type(4))) int   v4i;

#define EPS 0.001f

#define AS1 __attribute__((address_space(1)))
#define AS3 __attribute__((address_space(3)))

#if defined(__HIP_DEVICE_COMPILE__) && __has_builtin(__builtin_amdgcn_global_load_async_to_lds_b128)
#define HAS_ASYNC_LDS 1
#else
#define HAS_ASYNC_LDS 0
#endif

#if HAS_ASYNC_LDS
// Generic LDS pointer's low 32 bits are the LDS offset (ISA §10.2 aperture
// mapping); AS1 global pointers are bit-identical
// to generic global pointers. The async builtin takes v4i-typed AS pointers.
static __device__ __forceinline__ AS3 v4i* to_lds_v4(void* p) {
    return (AS3 v4i*)(uint32_t)(uintptr_t)p;
}
static __device__ __forceinline__ AS1 v4i* to_gbl_v4(const void* p) {
    return (AS1 v4i*)(uintptr_t)p;
}
static __device__ __forceinline__ void wait_async0() {
#if __has_builtin(__builtin_amdgcn_s_wait_asynccnt)
    __builtin_amdgcn_s_wait_asynccnt(0);
#else
    asm volatile("s_wait_asynccnt 0x0" ::: "memory");
#endif
}
#endif

// ---------------------------------------------------------------------------
// Kernel 1: KNN (top-16 nearest neighbors per point, excluding self).
// One block (128 threads) per batch; points staged in LDS. Neighbor order is
// irrelevant downstream (mean pool over K): tie-safe iterative min-extraction
// with (dist, index) lexicographic keys.
// ---------------------------------------------------------------------------
__global__ void __launch_bounds__(128) knn_topk_kernel(
    const float* __restrict__ pts,  // (B,128,2)
    int* __restrict__ knn)          // (B,128,16)
{
    __shared__ float px[128];
    __shared__ float py[128];
    const int b = blockIdx.x;
    const int t = threadIdx.x;

    const float* p = pts + (size_t)b * 128 * 2;
    px[t] = p[t * 2 + 0];
    py[t] = p[t * 2 + 1];
    __syncthreads();

    const float qx = px[t];
    const float qy = py[t];

    int* o = knn + ((size_t)b * 128 + t) * 16;

    float prevd = -1.0f;
    int   prevm = -1;
    for (int j = 0; j < 16; ++j) {
        float bd = 3.4e38f;
        int   bm = 128;
        for (int m = 0; m < 128; ++m) {
            if (m == t) continue;
            const float dx = qx - px[m];
            const float dy = qy - py[m];
            const float dd = dx * dx + dy * dy;
            const bool gt = (dd > prevd) || (dd == prevd && m > prevm);
            if (gt && (dd < bd || (dd == bd && m < bm))) { bd = dd; bm = m; }
        }
        o[j] = bm;
        prevd = bd;
        prevm = bm;
    }
}

// ---------------------------------------------------------------------------
// Kernel 1b: one-shot weight transpose (cin x 64) -> (64 x cin) so each
// WMMA lane's K-pair of the B operand is contiguous (b64 loads).
// ---------------------------------------------------------------------------
template <int CIN>
__global__ void __launch_bounds__(256) transpose_w_kernel(
    const float* __restrict__ src,  // (CIN, 64)
    float* __restrict__ dst)        // (64, CIN)
{
    const int t = blockIdx.x * 256 + threadIdx.x;  // over CIN*64
    const int c = t & (CIN - 1);
    const int d = t / CIN;
    dst[d * CIN + c] = src[c * 64 + d];
}

// ---------------------------------------------------------------------------
// WMMA helper: C[16x16] += A[16xCIN] * W^T tile. A in LDS row-major (stride
// CIN). Wtp = Wt + d*CIN (transposed weights, row = output channel d).
// f32 WMMA 16x16x4 operand layout:
//   A: lane L -> M = L%16; VGPR0/1 hold K = 2*(L/16), 2*(L/16)+1
//   B: lane L -> N = L%16; VGPR0/1 hold K rows 2*(L/16), 2*(L/16)+1
// ---------------------------------------------------------------------------
template <int CIN>
static __device__ __forceinline__ v8f gemm_tile(const float* __restrict__ As,
                                                const float* __restrict__ Wtp,
                                                int lane)
{
    const int half = lane >> 4;
    const int l16  = lane & 15;
    v8f acc = {0.f, 0.f, 0.f, 0.f, 0.f, 0.f, 0.f, 0.f};
#pragma unroll
    for (int kk = 0; kk < CIN / 4; ++kk) {
        const int c0 = kk * 4 + half * 2;
        v2f a;
        a.x = As[l16 * CIN + c0];
        a.y = As[l16 * CIN + c0 + 1];
        const v2f bb = *reinterpret_cast<const v2f*>(Wtp + c0);  // b64 load
        acc = __builtin_amdgcn_wmma_f32_16x16x4_f32(
            /*neg_a=*/false, a, /*neg_b=*/false, bb,
            /*c_mod=*/(short)0, acc, /*reuse_a=*/false, /*reuse_b=*/false);
    }
    return acc;
}

static __device__ __forceinline__ void bn_relu_to_lds(
    v8f acc,
    const float* __restrict__ g, const float* __restrict__ be,
    const float* __restrict__ mu, const float* __restrict__ va,
    int d, int half, float* __restrict__ dstCol /* LDS base + column d */)
{
    const float scale = g[d] * rsqrtf(va[d] + EPS);
    const float shift = be[d] - mu[d] * scale;
#pragma unroll
    for (int j = 0; j < 8; ++j) {
        float x = acc[j] * scale + shift;
        x = x > 0.f ? x : 0.f;
        dstCol[(j + 8 * half) * 64] = x;  // row = j + 8*half (C/D layout)
    }
}

// ---------------------------------------------------------------------------
// Kernel 2: fused edge-feature gather + 3-layer MLP + mean pool + shortcut.
// One block (128 threads = 4 waves) per (b, n). Each wave owns a 16-channel
// output tile; M = 16 neighbor rows.
// ---------------------------------------------------------------------------
__global__ void __launch_bounds__(128) edgeconv_fused_kernel(
    const float* __restrict__ feat,   // (B,128,64)
    const int*   __restrict__ knn,    // (B,128,16)
    const float* __restrict__ wt0, const float* __restrict__ g0,
    const float* __restrict__ b0, const float* __restrict__ m0,
    const float* __restrict__ v0,
    const float* __restrict__ wt1, const float* __restrict__ g1,
    const float* __restrict__ b1, const float* __restrict__ m1,
    const float* __restrict__ v1,
    const float* __restrict__ wt2, const float* __restrict__ g2,
    const float* __restrict__ b2, const float* __restrict__ m2,
    const float* __restrict__ v2p,
    const float* __restrict__ wtsc, const float* __restrict__ gsc,
    const float* __restrict__ bsc, const float* __restrict__ msc,
    const float* __restrict__ vsc,
    float* __restrict__ out)          // (B,128,64)
{
    __shared__ float ctr[64];        // center feature vector
    __shared__ int   nidx[16];       // neighbor indices
    __shared__ float NB[16 * 64];    // 4 KB gathered neighbor rows
    __shared__ float A0[16 * 128];   // 8 KB edge-feature tile / layer1 output
    __shared__ float A1[16 * 64];    // 4 KB layer0 output

    const int t   = threadIdx.x;
    const int blk = blockIdx.x;      // b*128 + n
    const int n   = blk & 127;
    const float* fbase = feat + (size_t)(blk >> 7) * (128 * 64);

    const int lane = t & 31;
    const int w    = t >> 5;          // wave id: output channel tile
    const int half = lane >> 4;
    const int l16  = lane & 15;
    const int d    = w * 16 + l16;    // output channel this lane owns

    // Warm L2->L0 path for this lane's weight rows (global_prefetch_b8).
    __builtin_prefetch(wt0 + d * 128, 0, 1);
    __builtin_prefetch(wt1 + d * 64, 0, 1);
    __builtin_prefetch(wt2 + d * 64, 0, 1);

    if (t < 64) ctr[t] = fbase[n * 64 + t];
    if (t < 16) nidx[t] = knn[(size_t)blk * 16 + t];
    __syncthreads();

#if HAS_ASYNC_LDS
    // Async DMA the 16 gathered neighbor rows straight into LDS
    // (global_load_async_to_lds_b128, tracked by ASYNCcnt).
    {
#pragma unroll
        for (int e = t; e < 16 * 16; e += 128) {   // float4 granularity
            const int r  = e >> 4;
            const int c4 = e & 15;
            const float* gsrc = fbase + nidx[r] * 64 + c4 * 4;
            __builtin_amdgcn_global_load_async_to_lds_b128(
                to_gbl_v4(gsrc), to_lds_v4(&NB[r * 64 + c4 * 4]), 0, 0);
        }
        wait_async0();
    }
    __syncthreads();
    // Build A0 = [center | nbr - center] from staged rows.
#pragma unroll
    for (int e = t; e < 16 * 64; e += 128) {
        const int r = e >> 6;
        const int c = e & 63;
        const float cv = ctr[c];
        A0[r * 128 + c]      = cv;
        A0[r * 128 + 64 + c] = NB[r * 64 + c] - cv;
    }
#else
    // Fallback: gather through VGPRs.
#pragma unroll
    for (int e = t; e < 16 * 64; e += 128) {
        const int r = e >> 6;
        const int c = e & 63;
        const float fv = fbase[nidx[r] * 64 + c];
        const float cv = ctr[c];
        A0[r * 128 + c]      = cv;
        A0[r * 128 + 64 + c] = fv - cv;
    }
#endif
    __syncthreads();

    // Layer 0: 16x128 @ 128x64
    v8f acc = gemm_tile<128>(A0, wt0 + d * 128, lane);
    bn_relu_to_lds(acc, g0, b0, m0, v0, d, half, A1 + d);
    __syncthreads();

    // Layer 1: 16x64 @ 64x64 (output reuses A0 space; safe after barrier)
    acc = gemm_tile<64>(A1, wt1 + d * 64, lane);
    bn_relu_to_lds(acc, g1, b1, m1, v1, d, half, A0 + d);
    __syncthreads();

    // Layer 2: 16x64 @ 64x64, BN+ReLU, mean-pool over the 16 rows in-register
    acc = gemm_tile<64>(A0, wt2 + d * 64, lane);
    {
        const float scale = g2[d] * rsqrtf(v2p[d] + EPS);
        const float shift = b2[d] - m2[d] * scale;
        float part = 0.f;
#pragma unroll
        for (int j = 0; j < 8; ++j) {
            float x = acc[j] * scale + shift;
            part += (x > 0.f ? x : 0.f);
        }
        // lanes L and L^16 hold M=0..7 and M=8..15 of the same channel
        part += __shfl_xor(part, 16, 32);
        const float pooled = part * (1.0f / 16.0f);

        // Shortcut: BN(features[b,n,:] @ wsc[:,d]) using transposed row.
        const float* wrow = wtsc + d * 64;
        float sc = 0.f;
#pragma unroll
        for (int c = 0; c < 64; c += 4) {
            const float4 wv = *reinterpret_cast<const float4*>(wrow + c);
            sc += ctr[c + 0] * wv.x + ctr[c + 1] * wv.y +
                  ctr[c + 2] * wv.z + ctr[c + 3] * wv.w;
        }
        const float s2 = gsc[d] * rsqrtf(vsc[d] + EPS);
        sc = (sc - msc[d]) * s2 + bsc[d];

        if (half == 0) {
            float r = sc + pooled;
            out[(size_t)blk * 64 + d] = r > 0.f ? r : 0.f;
        }
    }
}

// ---------------------------------------------------------------------------
// Launch. Inputs in setup_inputs() order:
// 0 points, 1 features, 2..6 w0 g0 b0 m0 v0, 7..11 w1..., 12..16 w2...,
// 17..21 wsc gsc bsc msc vsc
// Workspace layout: [0, 2MB) knn indices; then transposed weights.
// ---------------------------------------------------------------------------
extern "C" void kernel_launch(void* const* d_in, const int* in_sizes, int n_in,
                              void* d_out, int out_size, void* d_ws, size_t ws_size,
                              hipStream_t stream)
{
    const float* points = (const float*)d_in[0];
    const float* feats  = (const float*)d_in[1];
    const float* w0 = (const float*)d_in[2];
    const float* g0 = (const float*)d_in[3];
    const float* b0 = (const float*)d_in[4];
    const float* m0 = (const float*)d_in[5];
    const float* v0 = (const float*)d_in[6];
    const float* w1 = (const float*)d_in[7];
    const float* g1 = (const float*)d_in[8];
    const float* b1 = (const float*)d_in[9];
    const float* m1 = (const float*)d_in[10];
    const float* v1 = (const float*)d_in[11];
    const float* w2 = (const float*)d_in[12];
    const float* g2 = (const float*)d_in[13];
    const float* b2 = (const float*)d_in[14];
    const float* m2 = (const float*)d_in[15];
    const float* v2 = (const float*)d_in[16];
    const float* wsc = (const float*)d_in[17];
    const float* gsc = (const float*)d_in[18];
    const float* bsc = (const float*)d_in[19];
    const float* msc = (const float*)d_in[20];
    const float* vsc = (const float*)d_in[21];

    const int B = 256, N = 128;

    int*   knn  = (int*)d_ws;                                   // 2 MB
    float* wt0  = (float*)((char*)d_ws + (size_t)B * N * 16 * 4);
    float* wt1  = wt0 + 128 * 64;
    float* wt2  = wt1 + 64 * 64;
    float* wtsc = wt2 + 64 * 64;

    knn_topk_kernel<<<B, N, 0, stream>>>(points, knn);
    transpose_w_kernel<128><<<(128 * 64) / 256, 256, 0, stream>>>(w0, wt0);
    transpose_w_kernel<64><<<(64 * 64) / 256, 256, 0, stream>>>(w1, wt1);
    transpose_w_kernel<64><<<(64 * 64) / 256, 256, 0, stream>>>(w2, wt2);
    transpose_w_kernel<64><<<(64 * 64) / 256, 256, 0, stream>>>(wsc, wtsc);

    edgeconv_fused_kernel<<<B * N, 128, 0, stream>>>(
        feats, knn,
        wt0, g0, b0, m0, v0,
        wt1, g1, b1, m1, v1,
        wt2, g2, b2, m2, v2,
        wtsc, gsc, bsc, msc, vsc,
        (float*)d_out);
}